// LengthRegulator_50414326120823
// MI455X (gfx1250) — compile-verified
//
#include <hip/hip_runtime.h>
#include <hip/hip_bf16.h>
#include <stdint.h>

// Shapes from the reference
#define B_DIM   16
#define T_IN    512
#define H_DIM   256
#define T_MEL   4096
#define HV4     (H_DIM / 4)   // 64 float4 per H-row
#define TILE_T  64            // time rows per block
#define NTHREADS 256

typedef float v4f __attribute__((ext_vector_type(4)));

// CDNA5 async global->LDS path (guarded; falls back to plain LDS staging)
#if defined(__gfx1250__) && __has_builtin(__builtin_amdgcn_global_load_async_to_lds_b32) && __has_builtin(__builtin_amdgcn_s_wait_asynccnt)
#define USE_ASYNC_LDS 1
#else
#define USE_ASYNC_LDS 0
#endif

// -----------------------------------------------------------------------------
// Gather kernel: out[b,t,:] = dur==0 ? 0 : x[b, dur-1, :]
// Block = (b, 64-row tile of T_MEL). 256 threads, 8 waves.
// Each wave's 32 lanes cover one contiguous 512B half-row -> coalesced b128,
// and the gather index is wave-uniform.
// -----------------------------------------------------------------------------
__global__ __launch_bounds__(NTHREADS)
void lr_gather_kernel(const float* __restrict__ x,
                      const int* __restrict__ duration,
                      float* __restrict__ out) {
    __shared__ int s_dur[TILE_T];

    const int tid = threadIdx.x;
    const int b  = blockIdx.x >> 6;          // blockIdx.x / (T_MEL/TILE_T)
    const int t0 = (blockIdx.x & 63) << 6;   // tile start in T_MEL

    const int* __restrict__ dur_row = duration + (size_t)b * T_MEL + t0;

    // Stage this tile's 64 duration indices into LDS.
#if USE_ASYNC_LDS
    if (tid < TILE_T) {
        __builtin_amdgcn_global_load_async_to_lds_b32(
            (__attribute__((address_space(1))) int*)(dur_row + tid),
            (__attribute__((address_space(3))) int*)(&s_dur[tid]),
            /*offset=*/0, /*cpol=*/0);
        __builtin_amdgcn_s_wait_asynccnt(0);
    }
    __syncthreads();
#else
    if (tid < TILE_T) {
        s_dur[tid] = dur_row[tid];
    }
    __syncthreads();
#endif

    const int c  = tid & (HV4 - 1);  // float4 column within the H-row (0..63)
    const int rg = tid >> 6;         // row group 0..3 (rows strided by 4)

    const v4f* __restrict__ xb =
        (const v4f*)(x + (size_t)b * T_IN * H_DIM);            // [T_IN][HV4]
    v4f* __restrict__ ob =
        (v4f*)(out + ((size_t)b * T_MEL + t0) * H_DIM);        // [TILE_T][HV4]

    #pragma unroll
    for (int i = 0; i < TILE_T / 4; ++i) {
        const int r = rg + (i << 2);          // 0..63, wave-uniform index read
        const int d = s_dur[r];
        v4f v = {0.f, 0.f, 0.f, 0.f};
        if (d > 0) {
            v = xb[(size_t)(d - 1) * HV4 + c];   // global_load_b128, L2-resident
        }
        // Streaming, write-once output: non-temporal store (keep x hot in L2)
        __builtin_nontemporal_store(v, &ob[(size_t)r * HV4 + c]);
    }
}

// -----------------------------------------------------------------------------
// mel_len kernel: first index of 0 in duration[b,:], else T_MEL.
// One block per batch row; strided scan + LDS min-tree.
// Written as float (tuple outputs are concatenated into the fp32 d_out).
// -----------------------------------------------------------------------------
__global__ __launch_bounds__(NTHREADS)
void lr_mellen_kernel(const int* __restrict__ duration,
                      float* __restrict__ mel_out) {
    __shared__ int red[NTHREADS];

    const int b = blockIdx.x;
    const int* __restrict__ drow = duration + (size_t)b * T_MEL;

    int m = T_MEL;
    for (int t = threadIdx.x; t < T_MEL; t += NTHREADS) {
        if (drow[t] == 0) m = min(m, t);
    }
    red[threadIdx.x] = m;
    __syncthreads();

    #pragma unroll
    for (int s = NTHREADS / 2; s > 0; s >>= 1) {
        if (threadIdx.x < (unsigned)s) {
            red[threadIdx.x] = min(red[threadIdx.x], red[threadIdx.x + s]);
        }
        __syncthreads();
    }

    if (threadIdx.x == 0) {
        mel_out[b] = (float)red[0];
    }
}

// -----------------------------------------------------------------------------
// Launch
// -----------------------------------------------------------------------------
extern "C" void kernel_launch(void* const* d_in, const int* in_sizes, int n_in,
                              void* d_out, int out_size, void* d_ws, size_t ws_size,
                              hipStream_t stream) {
    (void)in_sizes; (void)n_in; (void)out_size; (void)d_ws; (void)ws_size;

    const float* x        = (const float*)d_in[0];   // [B, T_IN, H] fp32
    const int*   duration = (const int*)d_in[1];     // [B, T_MEL] int

    float* out     = (float*)d_out;                          // [B, T_MEL, H]
    float* mel_out = out + (size_t)B_DIM * T_MEL * H_DIM;    // [B]

    dim3 gather_grid(B_DIM * (T_MEL / TILE_T));  // 1024 blocks
    lr_gather_kernel<<<gather_grid, NTHREADS, 0, stream>>>(x, duration, out);

    lr_mellen_kernel<<<B_DIM, NTHREADS, 0, stream>>>(duration, mel_out);
}